// FlotEncoder_67327907332133
// MI455X (gfx1250) — compile-verified
//
#include <hip/hip_runtime.h>

// ---------------------------------------------------------------------------
// FLOT SetConv encoder for MI455X (gfx1250, wave32, WMMA + TDM).
//
// Roofline: ~35 GFLOP of matmul vs ~1.6 GB of activation traffic at
// 23.3 TB/s -> memory/gather bound. We keep full fp32 precision and use
// V_WMMA_F32_16X16X4_F32 (fp32 matrix pipe) with A tiles staged in LDS.
// Weight tiles are streamed into LDS by the Tensor Data Mover
// (tensor_load_to_lds) with double buffering: tile nt+1 is in flight while
// tile nt is consumed, synchronized via s_wait_tensorcnt (in-order TENSORcnt).
// GroupNorm stats use deterministic fixed-tree reductions (no fp atomics) so
// graph replays are bitwise reproducible.
// ---------------------------------------------------------------------------

typedef __attribute__((ext_vector_type(2))) float v2f;
typedef __attribute__((ext_vector_type(8))) float v8f;
typedef __attribute__((ext_vector_type(4))) unsigned int u32x4;
typedef __attribute__((ext_vector_type(8))) int i32x8;
typedef __attribute__((ext_vector_type(4))) int i32x4;

#define BB 4
#define NP 4096
#define KN 32
#define GG 8
#define EPSI 1e-5f
#define NSLOPE 0.1f

#if defined(__gfx1250__) && __has_builtin(__builtin_amdgcn_tensor_load_to_lds)
#define HAVE_TDM 1
#else
#define HAVE_TDM 0
#endif

// ---------------------------------------------------------------------------
// wave-wide 64-bit min reduction (wave32) via 32-bit shuffles
// ---------------------------------------------------------------------------
__device__ __forceinline__ unsigned long long wave_min_u64(unsigned long long v) {
#pragma unroll
  for (int m = 16; m; m >>= 1) {
    unsigned lo = (unsigned)v, hi = (unsigned)(v >> 32);
    lo = __shfl_xor(lo, m, 32);
    hi = __shfl_xor(hi, m, 32);
    unsigned long long o = ((unsigned long long)hi << 32) | (unsigned long long)lo;
    if (o < v) v = o;
  }
  return v;
}

#if HAVE_TDM
// ---------------------------------------------------------------------------
// TDM: load a (C1 rows x 16 cols) fp32 tile of W (row stride Cout elements)
// into LDS at byte offset lds_off. D# packing per CDNA5 ISA ch.8.
// This toolchain exposes the 6-arg builtin: (g0, g1, g2, g3, g4, cpol);
// groups 2..4 are unused for a 2-D tensor -> zeros.
// ---------------------------------------------------------------------------
__device__ __forceinline__ void tdm_load_wtile(const float* gptr, unsigned lds_off,
                                               int C1, int Cout) {
  unsigned long long ga = (unsigned long long)(uintptr_t)gptr;
  u32x4 g0;
  g0[0] = 1u;                                  // count=1, user mode
  g0[1] = lds_off;                             // lds_addr (bytes)
  g0[2] = (unsigned)ga;                        // global_addr[31:0]
  g0[3] = (unsigned)(ga >> 32) | (2u << 30);   // global_addr[56:32] | type=2
  i32x8 g1;
  g1[0] = (int)(2u << 16);                     // data_size=2 (4 bytes)
  g1[1] = (int)(((unsigned)Cout & 0xffffu) << 16);  // tensor_dim0[15:0]
  g1[2] = (int)(((unsigned)C1 & 0xffffu) << 16);    // dim0 hi=0 | tensor_dim1[15:0]
  g1[3] = (int)(16u << 16);                    // dim1 hi=0 | tile_dim0=16
  g1[4] = (int)((unsigned)C1 & 0xffffu);       // tile_dim1=C1, tile_dim2=0
  g1[5] = (int)(unsigned)Cout;                 // tensor_dim0_stride[31:0]
  g1[6] = 0;                                   // stride hi / dim1_stride lo
  g1[7] = 0;
  i32x4 z4 = {0, 0, 0, 0};                     // 2-D tensor: groups 2/3 unused
  i32x8 z8 = {0, 0, 0, 0, 0, 0, 0, 0};
  __builtin_amdgcn_tensor_load_to_lds(g0, g1, z4, z4, z8, 0);
}
#endif

// ---------------------------------------------------------------------------
// kNN: one wave per query point; pc[b] staged in LDS (NP*3*4 = 48KB).
// 32 passes of "min key strictly greater than last", key = (distbits<<32)|idx,
// which reproduces jax.lax.top_k ordering (ascending dist, ties by index).
// ---------------------------------------------------------------------------
__global__ void knn_kernel(const float* __restrict__ pc,
                           int* __restrict__ nbrs, float* __restrict__ edge) {
  extern __shared__ float pcs[]; // NP*3
  const int tid = threadIdx.x, lane = tid & 31, wave = tid >> 5;
  const int qid = blockIdx.x * 4 + wave;       // 4 waves per block, same batch
  const int b = qid / NP, n = qid % NP;
  for (int i = tid; i < NP * 3; i += blockDim.x)
    pcs[i] = pc[(size_t)b * NP * 3 + i];
  __syncthreads();
  const float qx = pcs[n * 3 + 0], qy = pcs[n * 3 + 1], qz = pcs[n * 3 + 2];
  unsigned long long last = 0ull;
  for (int sel = 0; sel < KN; ++sel) {
    unsigned long long best = ~0ull;
    for (int j = lane; j < NP; j += 32) {
      float dx = pcs[j * 3 + 0] - qx;
      float dy = pcs[j * 3 + 1] - qy;
      float dz = pcs[j * 3 + 2] - qz;
      float d = dx * dx + dy * dy + dz * dz;
      unsigned long long key =
          ((unsigned long long)__float_as_uint(d) << 32) | (unsigned)j;
      if ((sel == 0 || key > last) && key < best) best = key;
    }
    best = wave_min_u64(best);
    last = best;
    if (lane == 0) {
      int idx = (int)(unsigned)best;
      int o = qid * KN + sel;
      nbrs[o] = idx;
      edge[o * 3 + 0] = pcs[idx * 3 + 0] - qx;
      edge[o * 3 + 1] = pcs[idx * 3 + 1] - qy;
      edge[o * 3 + 2] = pcs[idx * 3 + 2] - qz;
    }
  }
}

// ---------------------------------------------------------------------------
// x0 = concat(pc, fea) -> [B, N, 6]
// ---------------------------------------------------------------------------
__global__ void concat_kernel(const float* __restrict__ pc,
                              const float* __restrict__ fea,
                              float* __restrict__ sig0) {
  long long total = (long long)BB * NP * 6;
  for (long long idx = (long long)blockIdx.x * blockDim.x + threadIdx.x;
       idx < total; idx += (long long)gridDim.x * blockDim.x) {
    int c = (int)(idx % 6);
    long long bn = idx / 6;
    sig0[idx] = (c < 3) ? pc[bn * 3 + c] : fea[bn * 3 + c - 3];
  }
}

// ---------------------------------------------------------------------------
// GEMM: Out[M,Cout] = A[M,C1] * W[C1,Cout] + bias, via v_wmma_f32_16x16x4_f32.
// blockDim = 128 (4 waves); each wave owns a 16-row M-tile.
// GATHER mode builds A rows as concat(edge_feats[row], signal[b, nbr[row], :]).
// A tiles staged in LDS per wave (allows in-place GEMM: Out may alias Asrc).
// W tiles are double-buffered in LDS and streamed by the TDM (wave 0 issues
// tensor_load_to_lds for tile nt+1 while tile nt is consumed; s_wait_tensorcnt
// provides the producer/consumer ordering, barriers publish to the other waves).
// Fragment layouts per CDNA5 ISA tables: A 16x4 fp32: lanes 0-15 hold K=k,k+1,
// lanes 16-31 hold K=k+2,k+3 (row = lane&15). D: VGPR i -> row i (+8 for upper
// half-lanes), col = lane&15. B assumed symmetric to A.
// ---------------------------------------------------------------------------
template <bool GATHER>
__global__ void gemm_kernel(const float* __restrict__ Asrc,
                            const int* __restrict__ nbrs,
                            const float* __restrict__ edge,
                            const float* __restrict__ W,
                            const float* __restrict__ bias,
                            float* __restrict__ Out,
                            int C1, int Kpad, int Cout, int Csig, int rowsPerB) {
  extern __shared__ float sm[];
  float* sA = sm;                         // 4 * 16 * Kpad
  float* sW0 = sm + 4 * 16 * Kpad;        // Kpad * 16 (tile buffer 0)
  float* sW1 = sW0 + Kpad * 16;           // Kpad * 16 (tile buffer 1)
  float* sBias = sW1 + Kpad * 16;         // Cout
  const int tid = threadIdx.x, lane = tid & 31, wave = tid >> 5;
  const int half = lane >> 4, r16 = lane & 15;
  const int m0 = blockIdx.x * 64 + wave * 16;
  const int nTiles = Cout >> 4;

  if (tid < Cout) sBias[tid] = bias[tid];

  // --- per-wave A tile staging (gather/concat or plain rows) ---
  float* sAw = sA + wave * 16 * Kpad;
  for (int idx = lane; idx < 16 * Kpad; idx += 32) {
    int r = idx / Kpad, k = idx - r * Kpad;
    float v = 0.f;
    if (k < C1) {
      int row = m0 + r;
      if (GATHER) {
        if (k < 3) {
          v = edge[(size_t)row * 3 + k];
        } else {
          int nb = nbrs[row];
          int bb = row / rowsPerB;
          v = Asrc[((size_t)bb * NP + nb) * Csig + (k - 3)];
        }
      } else {
        v = Asrc[(size_t)row * C1 + k];
      }
    }
    sAw[idx] = v;
  }

#if HAVE_TDM
  // Zero the K-padding rows of both W buffers once; TDM only writes C1 rows.
  for (int idx = tid; idx < (Kpad - C1) * 16; idx += 128) {
    sW0[C1 * 16 + idx] = 0.f;
    sW1[C1 * 16 + idx] = 0.f;
  }
  const unsigned ldsWBase = (unsigned)__builtin_amdgcn_groupstaticsize() +
                            (unsigned)(4 * 16 * Kpad) * 4u;
  const unsigned wBytes = (unsigned)(Kpad * 16) * 4u;
  if (wave == 0) tdm_load_wtile(W, ldsWBase, C1, Cout); // prefetch tile 0
#endif

  __syncthreads();

  for (int nt = 0; nt < nTiles; ++nt) {
    float* sWt = (nt & 1) ? sW1 : sW0;
#if HAVE_TDM
    if (wave == 0) {
      if (nt + 1 < nTiles) {
        tdm_load_wtile(W + (size_t)(nt + 1) * 16,
                       ldsWBase + (unsigned)((nt + 1) & 1) * wBytes, C1, Cout);
        __builtin_amdgcn_s_wait_tensorcnt(1); // tile nt done, nt+1 in flight
      } else {
        __builtin_amdgcn_s_wait_tensorcnt(0);
      }
    }
    __syncthreads(); // publish TDM-written tile to all waves
#else
    for (int idx = tid; idx < Kpad * 16; idx += 128) {
      int k = idx >> 4, c = idx & 15;
      sWt[idx] = (k < C1) ? W[(size_t)k * Cout + nt * 16 + c] : 0.f;
    }
    __syncthreads();
#endif

    v8f acc = {};
    for (int k = 0; k < Kpad; k += 4) {
      v2f a, bf;
      a.x = sAw[r16 * Kpad + k + 2 * half];
      a.y = sAw[r16 * Kpad + k + 1 + 2 * half];
      bf.x = sWt[(k + 2 * half) * 16 + r16];
      bf.y = sWt[(k + 1 + 2 * half) * 16 + r16];
      acc = __builtin_amdgcn_wmma_f32_16x16x4_f32(false, a, false, bf,
                                                  (short)0, acc, false, false);
    }
    int c = nt * 16 + r16;
    float bv = sBias[c];
#pragma unroll
    for (int i = 0; i < 8; ++i) {
      int row = m0 + i + 8 * half;
      Out[(size_t)row * Cout + c] = acc[i] + bv;
    }
    __syncthreads(); // all reads of sWt done before it is refilled
  }
}

// ---------------------------------------------------------------------------
// GroupNorm stats: one WG per (b, group); deterministic strided sums + fixed
// LDS tree reduction -> mean, rsqrt(var+eps).
// ---------------------------------------------------------------------------
__global__ void stat_kernel(const float* __restrict__ Y, float* __restrict__ meanA,
                            float* __restrict__ rstdA, int rowsPerB, int C) {
  const int bg = blockIdx.x, b = bg / GG, g = bg % GG;
  const int Cg = C / GG;
  const long long cnt = (long long)rowsPerB * Cg;
  const float* base = Y + (size_t)b * rowsPerB * C;
  float s1 = 0.f, s2 = 0.f;
  for (long long i = threadIdx.x; i < cnt; i += 256) {
    int row = (int)(i / Cg);
    int cc = g * Cg + (int)(i - (long long)row * Cg);
    float v = base[(size_t)row * C + cc];
    s1 += v;
    s2 += v * v;
  }
  __shared__ float r1[256], r2[256];
  r1[threadIdx.x] = s1;
  r2[threadIdx.x] = s2;
  __syncthreads();
  for (int s = 128; s; s >>= 1) {
    if (threadIdx.x < s) {
      r1[threadIdx.x] += r1[threadIdx.x + s];
      r2[threadIdx.x] += r2[threadIdx.x + s];
    }
    __syncthreads();
  }
  if (threadIdx.x == 0) {
    float mean = r1[0] / (float)cnt;
    float var = r2[0] / (float)cnt - mean * mean;
    meanA[bg] = mean;
    rstdA[bg] = rsqrtf(var + EPSI);
  }
}

// ---------------------------------------------------------------------------
// In-place GroupNorm affine + LeakyReLU
// ---------------------------------------------------------------------------
__global__ void gn_act_kernel(float* __restrict__ Y, const float* __restrict__ mean,
                              const float* __restrict__ rstd,
                              const float* __restrict__ gamma,
                              const float* __restrict__ beta,
                              long long total, int C, int rowsPerB) {
  const int Cg = C / GG;
  for (long long idx = (long long)blockIdx.x * blockDim.x + threadIdx.x;
       idx < total; idx += (long long)gridDim.x * blockDim.x) {
    long long row = idx / C;
    int c = (int)(idx - row * C);
    int b = (int)(row / rowsPerB);
    int g = c / Cg;
    float v = (Y[idx] - mean[b * GG + g]) * rstd[b * GG + g] * gamma[c] + beta[c];
    Y[idx] = (v >= 0.f) ? v : NSLOPE * v;
  }
}

// ---------------------------------------------------------------------------
// Apply GN2+LeakyReLU on the fly and max-pool over K neighbors -> S[B,N,C]
// ---------------------------------------------------------------------------
__global__ void maxpool_kernel(const float* __restrict__ Y,
                               const float* __restrict__ mean,
                               const float* __restrict__ rstd,
                               const float* __restrict__ gamma,
                               const float* __restrict__ beta,
                               float* __restrict__ S, int C) {
  const int Cg = C / GG;
  long long total = (long long)BB * NP * C;
  for (long long idx = (long long)blockIdx.x * blockDim.x + threadIdx.x;
       idx < total; idx += (long long)gridDim.x * blockDim.x) {
    long long bn = idx / C;
    int c = (int)(idx - bn * C);
    int b = (int)(bn / NP);
    int g = c / Cg;
    float mu = mean[b * GG + g], rs = rstd[b * GG + g];
    float ga = gamma[c], be = beta[c];
    const float* base = Y + (size_t)(bn * KN) * C + c;
    float m = -3.4028235e38f;
    for (int kk = 0; kk < KN; ++kk) {
      float v = (base[(size_t)kk * C] - mu) * rs * ga + be;
      v = (v >= 0.f) ? v : NSLOPE * v;
      m = fmaxf(m, v);
    }
    S[idx] = m;
  }
}

// ---------------------------------------------------------------------------
// [B, N, 128] -> [B, 128, N]
// ---------------------------------------------------------------------------
__global__ void out_kernel(const float* __restrict__ X, float* __restrict__ out) {
  long long total = (long long)BB * 128 * NP;
  for (long long idx = (long long)blockIdx.x * blockDim.x + threadIdx.x;
       idx < total; idx += (long long)gridDim.x * blockDim.x) {
    int n = (int)(idx % NP);
    long long t = idx / NP;
    int c = (int)(t % 128);
    int b = (int)(t / 128);
    out[idx] = X[((size_t)b * NP + n) * 128 + c];
  }
}

// ---------------------------------------------------------------------------
// host-side launch sequence
// ---------------------------------------------------------------------------
extern "C" void kernel_launch(void* const* d_in, const int* in_sizes, int n_in,
                              void* d_out, int out_size, void* d_ws, size_t ws_size,
                              hipStream_t stream) {
  (void)in_sizes; (void)n_in; (void)out_size; (void)ws_size;
  const float* pc = (const float*)d_in[0];
  const float* fea = (const float*)d_in[1];
  // params dict order: l{0,1,2} x (w,b,g,be) for s=1,2,3
  auto P = [&](int l, int s, int which) -> const float* {
    return (const float*)d_in[2 + l * 12 + (s - 1) * 4 + which];
  };

  char* ws = (char*)d_ws;
  size_t off = 0;
  auto alloc = [&](size_t bytes) -> void* {
    void* p = ws + off;
    off += (bytes + 255) & ~(size_t)255;
    return p;
  };
  int* nbrs   = (int*)alloc((size_t)BB * NP * KN * 4);
  float* edge = (float*)alloc((size_t)BB * NP * KN * 3 * 4);
  float* sig0 = (float*)alloc((size_t)BB * NP * 6 * 4);
  float* Ybig = (float*)alloc((size_t)BB * NP * KN * 128 * 4); // stage1/2 acts
  float* Spool = (float*)alloc((size_t)BB * NP * 128 * 4);
  float* Ya   = (float*)alloc((size_t)BB * NP * 128 * 4);
  float* Yb   = (float*)alloc((size_t)BB * NP * 128 * 4);
  float* stats = (float*)alloc(6 * 32 * 4);

  knn_kernel<<<BB * NP / 4, 128, NP * 3 * 4, stream>>>(pc, nbrs, edge);
  concat_kernel<<<(BB * NP * 6 + 255) / 256, 256, 0, stream>>>(pc, fea, sig0);

  const int dims[3] = {32, 64, 128};
  const float* sig = sig0;
  int Cin = 6;
  const long long Mbig = (long long)BB * NP * KN;
  float* Y3bufs[2] = {Ya, Yb};

  for (int l = 0; l < 3; ++l) {
    int Cout = dims[l];
    float* m1 = stats + 0 * 32; float* r1 = stats + 1 * 32;
    float* m2 = stats + 2 * 32; float* r2 = stats + 3 * 32;
    float* m3 = stats + 4 * 32; float* r3 = stats + 5 * 32;

    // stage 1: gather-concat GEMM
    int C1 = Cin + 3, Kp1 = (C1 + 3) & ~3;
    size_t sh1 = (size_t)(96 * Kp1 + Cout) * 4;
    gemm_kernel<true><<<(int)(Mbig / 64), 128, sh1, stream>>>(
        sig, nbrs, edge, P(l, 1, 0), P(l, 1, 1), Ybig, C1, Kp1, Cout, Cin, NP * KN);
    stat_kernel<<<BB * GG, 256, 0, stream>>>(Ybig, m1, r1, NP * KN, Cout);
    gn_act_kernel<<<4096, 256, 0, stream>>>(Ybig, m1, r1, P(l, 1, 2), P(l, 1, 3),
                                            Mbig * Cout, Cout, NP * KN);

    // stage 2: plain GEMM (in place; rows buffered in LDS before overwrite)
    int Kp2 = Cout;
    size_t sh2 = (size_t)(96 * Kp2 + Cout) * 4;
    gemm_kernel<false><<<(int)(Mbig / 64), 128, sh2, stream>>>(
        Ybig, nullptr, nullptr, P(l, 2, 0), P(l, 2, 1), Ybig, Cout, Kp2, Cout, 0,
        NP * KN);
    stat_kernel<<<BB * GG, 256, 0, stream>>>(Ybig, m2, r2, NP * KN, Cout);
    maxpool_kernel<<<2048, 256, 0, stream>>>(Ybig, m2, r2, P(l, 2, 2), P(l, 2, 3),
                                             Spool, Cout);

    // stage 3: pooled GEMM
    float* Y3 = Y3bufs[l & 1];
    gemm_kernel<false><<<BB * NP / 64, 128, sh2, stream>>>(
        Spool, nullptr, nullptr, P(l, 3, 0), P(l, 3, 1), Y3, Cout, Cout, Cout, 0, NP);
    stat_kernel<<<BB * GG, 256, 0, stream>>>(Y3, m3, r3, NP, Cout);
    gn_act_kernel<<<512, 256, 0, stream>>>(Y3, m3, r3, P(l, 3, 2), P(l, 3, 3),
                                           (long long)BB * NP * Cout, Cout, NP);
    sig = Y3;
    Cin = Cout;
  }

  out_kernel<<<2048, 256, 0, stream>>>(sig, (float*)d_out);
}